// DropKANLayer_22574348107991
// MI455X (gfx1250) — compile-verified
//
#include <hip/hip_runtime.h>
#include <hip/hip_bf16.h>

// KAN layer as one fp32 GEMM on CDNA5 WMMA (v_wmma_f32_16x16x4_f32), with
// CDNA5 async global->LDS copies (ASYNCcnt) and double-buffered tiles.
// Workspace layout (floats):
//   A   : [1024][12288]  activations (spline basis + silu)   = 48 MB
//   W   : [12288][512]   packed weights (scale_sp*coef | sb) = 24 MB
//   P   : [8][1024][512] split-K partials                    = 16 MB
// Total ws requirement: ~88 MB.

#define IN_DIM  512
#define OUT_DIM 512
#define NUM     20
#define KORD    3
#define BATCH   1024
#define NKNOT   (NUM + 1 + 2 * KORD)   // 27 knots per input dim
#define NBASIS  (NUM + KORD)           // 23 spline basis functions
#define KPERI   (NBASIS + 1)           // 24 = 23 spline + 1 silu/base
#define KTOT    (IN_DIM * KPERI)       // 12288 = GEMM K

typedef __attribute__((ext_vector_type(2))) float v2f;
typedef __attribute__((ext_vector_type(4))) float v4f;
typedef __attribute__((ext_vector_type(8))) float v8f;

// ---------------------------------------------------------------------------
// Phase 1: per-(b,i) cubic B-spline basis (Cox-de Boor, K=3) + silu.
// ---------------------------------------------------------------------------
__global__ __launch_bounds__(256) void kan_act_kernel(const float* __restrict__ x,
                                                      const float* __restrict__ grid,
                                                      float* __restrict__ A) {
    const int t = blockIdx.x * 256 + threadIdx.x;     // 0 .. BATCH*IN_DIM-1
    const int b = t >> 9;                             // batch index
    const int i = t & (IN_DIM - 1);                   // input dim index

    const float xv = x[b * IN_DIM + i];

    float g[NKNOT];
#pragma unroll
    for (int j = 0; j < NKNOT; ++j) g[j] = grid[i * NKNOT + j];

    float B[NKNOT - 1];
#pragma unroll
    for (int j = 0; j < NKNOT - 1; ++j)
        B[j] = (xv >= g[j] && xv < g[j + 1]) ? 1.0f : 0.0f;

#pragma unroll
    for (int ord = 1; ord <= KORD; ++ord) {
#pragma unroll
        for (int j = 0; j < NKNOT - 1 - ord; ++j) {
            const float left  = (xv - g[j]) / (g[j + ord] - g[j]);
            const float right = (g[j + ord + 1] - xv) / (g[j + ord + 1] - g[j + 1]);
            B[j] = left * B[j] + right * B[j + 1];
        }
    }

    float* dst = A + (size_t)b * KTOT + i * KPERI;
#pragma unroll
    for (int k = 0; k < NBASIS; ++k) dst[k] = B[k];
    dst[NBASIS] = xv / (1.0f + __expf(-xv));          // silu(x)
}

// ---------------------------------------------------------------------------
// Phase 2: packed weight  W[(i*24+k)][o] = scale_sp[i,o]*coef[i,o,k],
//          W[(i*24+23)][o] = scale_base[i,o].
// ---------------------------------------------------------------------------
__global__ __launch_bounds__(256) void kan_weight_kernel(const float* __restrict__ coef,
                                                         const float* __restrict__ sb,
                                                         const float* __restrict__ ssp,
                                                         float* __restrict__ W) {
    const int t = blockIdx.x * 256 + threadIdx.x;     // 0 .. IN_DIM*OUT_DIM-1
    const int i = t >> 9;
    const int o = t & (OUT_DIM - 1);

    const float s = ssp[i * OUT_DIM + o];
    const float* c = coef + ((size_t)i * OUT_DIM + o) * NBASIS;
    float* w = W + (size_t)(i * KPERI) * OUT_DIM + o;

#pragma unroll
    for (int k = 0; k < NBASIS; ++k) w[(size_t)k * OUT_DIM] = s * c[k];
    w[(size_t)NBASIS * OUT_DIM] = sb[i * OUT_DIM + o];
}

// ---------------------------------------------------------------------------
// Phase 3: fp32 GEMM  P[z] = A[1024 x 1536-slice] @ W[slice x 512]
//   using v_wmma_f32_16x16x4_f32.
//   Block tile 128x64, BK=32, split-K=8, 8 wave32s; wave tile 64x16
//   => 4 v8f accumulators/wave, 32 wmma per K-step, double-buffered LDS
//   filled by GLOBAL_LOAD_ASYNC_TO_LDS_B128 (ASYNCcnt, no VGPR round-trip).
// ---------------------------------------------------------------------------
#define BM 128
#define BN 64
#define BK 32
#define SPLITK 8
#define KSPLIT (KTOT / SPLITK)   // 1536
#define LDA 36                   // 32+4 pad: conflict-free frags, 16B rows
#define LDB 72                   // 64+8 pad: 2*LDB == 16 (mod 64)

__device__ __forceinline__ void async_b128_to_lds(const float* gaddr, uint32_t lds_addr) {
    // GLOBAL_LOAD_ASYNC_TO_LDS_B128: each lane copies 16 bytes from the global address in vaddr to the LDS address in vdst,
    // tracked by ASYNCcnt. vdst holds the LDS byte address (low 32 bits of generic ptr).
    asm volatile("global_load_async_to_lds_b128 %0, %1, off"
                 :: "v"(lds_addr), "v"(gaddr) : "memory");
}

__global__ __launch_bounds__(256) void kan_gemm_kernel(const float* __restrict__ A,
                                                       const float* __restrict__ W,
                                                       float* __restrict__ P) {
    __shared__ __align__(16) float As[2][BM * LDA];   // 2 x 18432 B
    __shared__ __align__(16) float Bs[2][BK * LDB];   // 2 x  9216 B

    const int tid    = threadIdx.x;
    const int lane   = tid & 31;
    const int wv     = tid >> 5;      // wave 0..7
    const int laneLo = lane & 15;
    const int laneHi = lane >> 4;     // 0/1

    const int M0   = blockIdx.y * BM;
    const int N0   = blockIdx.x * BN;
    const int kbeg = blockIdx.z * KSPLIT;

    const int wm0 = (wv & 1) * 64;    // wave row offset within block tile
    const int wn0 = (wv >> 1) * 16;   // wave col offset within block tile

    v8f acc[4];
#pragma unroll
    for (int m = 0; m < 4; ++m) { v8f z = {}; acc[m] = z; }

    // Global->LDS tiling (b128 per lane):
    //  A tile 128x32 = 1024 xb128 -> 4 per thread; rows f>>3, col4 (f&7)*4
    //  B tile  32x64 =  512 xb128 -> 2 per thread; rows f>>4, col4 (f&15)*4
    const float* Abase = A + (size_t)M0 * KTOT + kbeg;
    const float* Wbase = W + (size_t)kbeg * OUT_DIM + N0;

    auto issue_tile = [&](int kb, int buf) {
#pragma unroll
        for (int j = 0; j < 4; ++j) {
            const int f    = tid + 256 * j;
            const int row  = f >> 3;
            const int col4 = (f & 7) * 4;
            async_b128_to_lds(Abase + (size_t)row * KTOT + kb + col4,
                              (uint32_t)(uintptr_t)&As[buf][row * LDA + col4]);
        }
#pragma unroll
        for (int j = 0; j < 2; ++j) {
            const int f    = tid + 256 * j;
            const int row  = f >> 4;
            const int col4 = (f & 15) * 4;
            async_b128_to_lds(Wbase + (size_t)(kb + row) * OUT_DIM + col4,
                              (uint32_t)(uintptr_t)&Bs[buf][row * LDB + col4]);
        }
    };

    issue_tile(0, 0);
    asm volatile("s_wait_asynccnt 0" ::: "memory");
    __syncthreads();

    int buf = 0;
    for (int kb = 0; kb < KSPLIT; kb += BK) {
        if (kb + BK < KSPLIT) issue_tile(kb + BK, buf ^ 1);

        const float* Asb = As[buf];
        const float* Bsb = Bs[buf];
#pragma unroll
        for (int ku = 0; ku < BK / 4; ++ku) {
            // B fragment (4x16, K x N): K = v + 2*laneHi, N = laneLo
            v2f bf;
            bf.x = Bsb[(ku * 4 + 0 + 2 * laneHi) * LDB + wn0 + laneLo];
            bf.y = Bsb[(ku * 4 + 1 + 2 * laneHi) * LDB + wn0 + laneLo];
#pragma unroll
            for (int tm = 0; tm < 4; ++tm) {
                // A fragment (16x4, M x K): M = laneLo, K = v + 2*laneHi
                const v2f af = *(const v2f*)(&Asb[(wm0 + tm * 16 + laneLo) * LDA +
                                                  ku * 4 + 2 * laneHi]);
                acc[tm] = __builtin_amdgcn_wmma_f32_16x16x4_f32(
                    /*neg_a=*/false, af, /*neg_b=*/false, bf,
                    /*c_mod=*/(short)0, acc[tm],
                    /*reuse_a=*/false, /*reuse_b=*/false);
            }
        }

        // Next tile fully resident + all waves done reading current buffer.
        asm volatile("s_wait_asynccnt 0" ::: "memory");
        __syncthreads();
        buf ^= 1;
    }

    // C/D layout: M = v + 8*laneHi, N = laneLo
    float* Pp = P + (size_t)blockIdx.z * (BATCH * OUT_DIM);
#pragma unroll
    for (int tm = 0; tm < 4; ++tm) {
        const int row = M0 + wm0 + tm * 16 + 8 * laneHi;
        const int col = N0 + wn0 + laneLo;
#pragma unroll
        for (int v = 0; v < 8; ++v)
            Pp[(size_t)(row + v) * OUT_DIM + col] = acc[tm][v];
    }
}

// ---------------------------------------------------------------------------
// Phase 4: deterministic split-K reduction.
// ---------------------------------------------------------------------------
__global__ __launch_bounds__(256) void kan_reduce_kernel(const float* __restrict__ P,
                                                         float* __restrict__ out) {
    const int e = blockIdx.x * 256 + threadIdx.x;     // 0 .. BATCH*OUT_DIM-1
    float s = 0.0f;
#pragma unroll
    for (int sk = 0; sk < SPLITK; ++sk)
        s += P[(size_t)sk * (BATCH * OUT_DIM) + e];
    out[e] = s;
}

// ---------------------------------------------------------------------------
extern "C" void kernel_launch(void* const* d_in, const int* in_sizes, int n_in,
                              void* d_out, int out_size, void* d_ws, size_t ws_size,
                              hipStream_t stream) {
    const float* x    = (const float*)d_in[0];   // (1024, 512)
    const float* grid = (const float*)d_in[1];   // (512, 27)
    const float* coef = (const float*)d_in[2];   // (512, 512, 23)
    const float* sb   = (const float*)d_in[3];   // (512, 512)
    const float* ssp  = (const float*)d_in[4];   // (512, 512)
    float* out = (float*)d_out;                  // (1024, 512)

    float* Aact = (float*)d_ws;                                  // 1024*12288
    float* Wext = Aact + (size_t)BATCH * KTOT;                   // 12288*512
    float* Part = Wext + (size_t)KTOT * OUT_DIM;                 // 8*1024*512

    kan_act_kernel<<<(BATCH * IN_DIM) / 256, 256, 0, stream>>>(x, grid, Aact);
    kan_weight_kernel<<<(IN_DIM * OUT_DIM) / 256, 256, 0, stream>>>(coef, sb, ssp, Wext);

    dim3 g(OUT_DIM / BN, BATCH / BM, SPLITK);    // 8 x 8 x 8 = 512 workgroups
    kan_gemm_kernel<<<g, 256, 0, stream>>>(Aact, Wext, Part);

    kan_reduce_kernel<<<(BATCH * OUT_DIM) / 256, 256, 0, stream>>>(Part, out);
}